// DenseMaxPooling_22883585753527
// MI455X (gfx1250) — compile-verified
//
#include <hip/hip_runtime.h>

// DenseMaxPooling over circular shifts {0, L/4, L/2, 3L/4}, L = 8192.
// Key identity: the rolled-column set for output column c is exactly the
// 4 columns congruent to (c mod L/4). So:
//   out[r, c] = max_k in[r, (c mod 2048) + 2048*k]
// => read each input element once, write each output element once.
// Pure streaming kernel: 128 MB in + 128 MB out = 256 MB @ 23.3 TB/s ~ 11 us.
// 128-bit vector VMEM ops + non-temporal hints (single-touch data, keep the
// 192 MB L2 clean).

typedef float v4f __attribute__((ext_vector_type(4)));

__global__ __launch_bounds__(256)
void dense_max_pool_kernel(const float* __restrict__ in,
                           float* __restrict__ out,
                           int rows) {
    constexpr int L = 8192;         // row length
    constexpr int Q = L / 4;        // 2048: quarter stride
    constexpr int GROUPS = Q / 4;   // 512 float4 groups per row

    const long tid = (long)blockIdx.x * blockDim.x + threadIdx.x;
    const long row = tid / GROUPS;
    const long g   = tid - row * GROUPS;
    if (row >= rows) return;

    const float* rp = in  + row * (long)L + g * 4;
    float*       wp = out + row * (long)L + g * 4;

    // Four coalesced 128-bit non-temporal loads (one per quarter).
    v4f a = __builtin_nontemporal_load((const v4f*)(rp));
    v4f b = __builtin_nontemporal_load((const v4f*)(rp + Q));
    v4f c = __builtin_nontemporal_load((const v4f*)(rp + 2 * Q));
    v4f d = __builtin_nontemporal_load((const v4f*)(rp + 3 * Q));

    // 3 v_max_num_f32 per lane-component (compiler may fuse into v_max3).
    v4f m;
#pragma unroll
    for (int i = 0; i < 4; ++i) {
        m[i] = fmaxf(fmaxf(a[i], b[i]), fmaxf(c[i], d[i]));
    }

    // Broadcast the reduced quarter to all four output quarters:
    // four coalesced 128-bit non-temporal stores.
    __builtin_nontemporal_store(m, (v4f*)(wp));
    __builtin_nontemporal_store(m, (v4f*)(wp + Q));
    __builtin_nontemporal_store(m, (v4f*)(wp + 2 * Q));
    __builtin_nontemporal_store(m, (v4f*)(wp + 3 * Q));
}

extern "C" void kernel_launch(void* const* d_in, const int* in_sizes, int n_in,
                              void* d_out, int out_size, void* d_ws, size_t ws_size,
                              hipStream_t stream) {
    (void)n_in; (void)d_ws; (void)ws_size; (void)out_size;

    const float* in = (const float*)d_in[0];
    float* out = (float*)d_out;

    constexpr int L = 8192;
    const int rows = in_sizes[0] / L;            // 4096 for the reference shape

    const long groups_per_row = L / 4 / 4;       // 512 float4 groups per row
    const long total_threads  = (long)rows * groups_per_row;
    const int  block = 256;                      // 8 wave32s per block
    const long grid  = (total_threads + block - 1) / block;

    dense_max_pool_kernel<<<(unsigned)grid, block, 0, stream>>>(in, out, rows);
}